// EfficientMemorySiLU_27848567947574
// MI455X (gfx1250) — compile-verified
//
#include <hip/hip_runtime.h>
#include <math.h>

typedef __attribute__((ext_vector_type(2))) float v2f;
typedef __attribute__((ext_vector_type(8))) float v8f;
typedef __attribute__((ext_vector_type(4))) unsigned int v4u;
typedef __attribute__((ext_vector_type(4))) int v4i;
typedef __attribute__((ext_vector_type(8))) int v8i;

#define DIMX   4096
#define NROWSX 8192
#define NTOTX  (NROWSX * DIMX)   // 33554432
#define RANKX  16
#define KSUBX  409               // int(4096 * 0.1)
#define NBINSX 4096
#define BKX    32                // K-tile staged to LDS per block

// scalar slots in ws (float indices)
#define S_ABSMAX  0   // uint bits
#define S_LO      1
#define S_HI      2
#define S_SUBAMAX 3   // uint bits
#define S_THR     4
#define S_SCALE   5
#define S_BASE    8   // unsigned long long at slots 8..9

#if defined(__HIP_DEVICE_COMPILE__) && __has_builtin(__builtin_amdgcn_tensor_load_to_lds)
#define HAVE_TDM 1
#else
#define HAVE_TDM 0
#endif

static __device__ __forceinline__ v8f wmma4(v2f a, v2f b, v8f c) {
  // D = A(16x4 f32) * B(4x16 f32) + C(16x16 f32)
  return __builtin_amdgcn_wmma_f32_16x16x4_f32(false, a, false, b, (short)0, c,
                                               false, false);
}

#if HAVE_TDM
// Issue a 2D TDM load: tile (tile_d1 rows x tile_d0 elems, fp32) from a
// row-major tensor with row stride `stride0` (elems) into LDS at lds_byte.
static __device__ __forceinline__ void tdm_load_2d(unsigned int lds_byte,
                                                   const void* gaddr,
                                                   unsigned int tile_d0,
                                                   unsigned int tile_d1,
                                                   unsigned int tensor_d0,
                                                   unsigned int tensor_d1,
                                                   unsigned int stride0) {
  unsigned long long ga = (unsigned long long)(size_t)gaddr;
  v4u g0;
  g0.x = 1u;                                       // count=1, user descriptor
  g0.y = lds_byte;                                 // lds_addr [63:32]
  g0.z = (unsigned int)(ga & 0xFFFFFFFFull);       // global_addr [95:64]
  g0.w = (unsigned int)((ga >> 32) & 0x01FFFFFFull) | 0x80000000u; // [120:96] | type=2
  v8i g1;
  g1[0] = 0x20000;                                 // data_size=2 (4 bytes)
  g1[1] = (int)((tensor_d0 & 0xFFFFu) << 16);      // tensor_dim0[15:0] @ bit48
  g1[2] = (int)((tensor_d0 >> 16) | ((tensor_d1 & 0xFFFFu) << 16));
  g1[3] = (int)((tensor_d1 >> 16) | (tile_d0 << 16)); // tile_dim0 @ bit112
  g1[4] = (int)tile_d1;                            // tile_dim1 (tile_dim2=0)
  g1[5] = (int)stride0;                            // tensor_dim0_stride[31:0]
  g1[6] = 0;
  g1[7] = 0;
  v4i z4 = {0, 0, 0, 0};
#if defined(__clang_major__) && (__clang_major__ >= 23)
  v8i z8 = {0, 0, 0, 0, 0, 0, 0, 0};
  __builtin_amdgcn_tensor_load_to_lds(g0, g1, z4, z4, z8, 0);
#else
  __builtin_amdgcn_tensor_load_to_lds(g0, g1, z4, z4, 0);
#endif
}
#endif

__global__ void k_zero(float* p, int n) {
  int i = blockIdx.x * blockDim.x + threadIdx.x;
  if (i < n) p[i] = 0.0f;
}

// Pass 1: silu -> out, per-column sum of squares (LDS ds_add_f32 then global),
// global abs-max (for histogram range).
__global__ void k_silu_stats(const float* __restrict__ x, float* __restrict__ out_res,
                             float* __restrict__ colnorm2, float* __restrict__ scal) {
  __shared__ float lcol[DIMX];
  __shared__ float lmax[256];
  int tid = threadIdx.x;
  for (int c = tid; c < DIMX; c += blockDim.x) lcol[c] = 0.0f;
  __syncthreads();
  int base = blockIdx.x * 65536;
  float m = 0.0f;
  for (int i = base + tid; i < base + 65536; i += blockDim.x) {
    float v = x[i];
    out_res[i] = v / (1.0f + expf(-v));
    atomicAdd(&lcol[i & (DIMX - 1)], v * v);
    m = fmaxf(m, fabsf(v));
  }
  lmax[tid] = m;
  __syncthreads();
  for (int s = blockDim.x / 2; s > 0; s >>= 1) {
    if (tid < s) lmax[tid] = fmaxf(lmax[tid], lmax[tid + s]);
    __syncthreads();
  }
  if (tid == 0)
    atomicMax((unsigned int*)&scal[S_ABSMAX], __float_as_uint(lmax[0]));
  for (int c = tid; c < DIMX; c += blockDim.x) atomicAdd(&colnorm2[c], lcol[c]);
}

// Single-block bitonic sort of 4096 column sum-squares; mark top-KSUBX columns.
__global__ void k_topk(const float* __restrict__ colnorm2, float* __restrict__ colmask) {
  __shared__ float s[DIMX];
  int tid = threadIdx.x;
  for (int i = tid; i < DIMX; i += blockDim.x) s[i] = colnorm2[i];
  for (int k = 2; k <= DIMX; k <<= 1)
    for (int j = k >> 1; j > 0; j >>= 1) {
      __syncthreads();
      for (int i = tid; i < DIMX; i += blockDim.x) {
        int ixj = i ^ j;
        if (ixj > i) {
          float a = s[i], b = s[ixj];
          bool up = ((i & k) == 0);
          if ((a > b) == up) { s[i] = b; s[ixj] = a; }
        }
      }
    }
  __syncthreads();
  float kth = s[DIMX - KSUBX];  // value of the KSUBX-th largest
  for (int i = tid; i < DIMX; i += blockDim.x)
    colmask[i] = (colnorm2[i] >= kth) ? 1.0f : 0.0f;
}

__global__ void k_prep1(float* scal) {
  if (threadIdx.x == 0 && blockIdx.x == 0) {
    float am = __uint_as_float(*(unsigned int*)&scal[S_ABSMAX]);
    scal[S_LO] = 0.0f;
    scal[S_HI] = fmaxf(am, 1e-30f) * 1.0000002f;
    *(unsigned long long*)&scal[S_BASE] = 0ull;
  }
}

// Histogram of |x| over [lo, hi) into 4096 bins (LDS accumulate then global).
__global__ void k_hist(const float* __restrict__ x, const float* __restrict__ scal,
                       unsigned int* __restrict__ hist) {
  __shared__ unsigned int lh[NBINSX];
  int tid = threadIdx.x;
  for (int i = tid; i < NBINSX; i += blockDim.x) lh[i] = 0u;
  __syncthreads();
  float lo = scal[S_LO], hi = scal[S_HI];
  float inv = (float)NBINSX / (hi - lo);
  int stride = gridDim.x * blockDim.x;
  for (int i = blockIdx.x * blockDim.x + tid; i < NTOTX; i += stride) {
    __builtin_prefetch(x + i + 4 * stride, 0, 0);
    float a = fabsf(x[i]);
    if (a >= lo && a < hi) {
      int b = (int)((a - lo) * inv);
      if (b > NBINSX - 1) b = NBINSX - 1;
      atomicAdd(&lh[b], 1u);
    }
  }
  __syncthreads();
  for (int i = tid; i < NBINSX; i += blockDim.x)
    if (lh[i]) atomicAdd(&hist[i], lh[i]);
}

// Find the histogram bin holding order-statistic floor(0.99*(N-1)).
// level 0: narrow [lo,hi] to that bin; level 1: interpolate final threshold.
__global__ void k_scan(const unsigned int* __restrict__ hist, float* __restrict__ scal,
                       int level, float* __restrict__ out_thr) {
  if (threadIdx.x != 0 || blockIdx.x != 0) return;
  const double p = 0.99 * (double)(NTOTX - 1);
  long long i0 = (long long)p;
  double frac = p - (double)i0;
  double lo = scal[S_LO], hi = scal[S_HI];
  double w = (hi - lo) / (double)NBINSX;
  unsigned long long cum = *(unsigned long long*)&scal[S_BASE];
  int b = NBINSX - 1;
  unsigned long long cumBefore = cum;
  unsigned int cb = 0;
  for (int i = 0; i < NBINSX; i++) {
    unsigned int c = hist[i];
    if (cum + c > (unsigned long long)i0) { b = i; cumBefore = cum; cb = c; break; }
    cum += c;
  }
  if (level == 0) {
    scal[S_LO] = (float)(lo + w * b);
    scal[S_HI] = (float)(lo + w * (b + 1));
    *(unsigned long long*)&scal[S_BASE] = cumBefore;
  } else {
    if (cb == 0) cb = 1;
    double pos = ((double)(i0 - (long long)cumBefore) + frac + 0.5) / (double)cb;
    double thr = lo + w * ((double)b + pos);
    scal[S_THR] = (float)thr;
    if (out_thr) *out_thr = (float)thr;
  }
}

// max |x_sub| over masked columns of the (non-outlier) residual
__global__ void k_subamax(const float* __restrict__ x, const float* __restrict__ colmask,
                          float* __restrict__ scal) {
  __shared__ float lm[256];
  float thr = scal[S_THR];
  float m = 0.0f;
  int stride = gridDim.x * blockDim.x;
  for (int i = blockIdx.x * blockDim.x + threadIdx.x; i < NTOTX; i += stride) {
    __builtin_prefetch(x + i + 4 * stride, 0, 0);
    float v = fabsf(x[i]);
    if (v <= thr && colmask[i & (DIMX - 1)] != 0.0f) m = fmaxf(m, v);
  }
  lm[threadIdx.x] = m;
  __syncthreads();
  for (int s = blockDim.x / 2; s > 0; s >>= 1) {
    if (threadIdx.x < s) lm[threadIdx.x] = fmaxf(lm[threadIdx.x], lm[threadIdx.x + s]);
    __syncthreads();
  }
  if (threadIdx.x == 0)
    atomicMax((unsigned int*)&scal[S_SUBAMAX], __float_as_uint(lm[0]));
}

__global__ void k_scale(float* scal) {
  if (threadIdx.x == 0 && blockIdx.x == 0) {
    float sm = __uint_as_float(*(unsigned int*)&scal[S_SUBAMAX]);
    scal[S_SCALE] = fmaxf(sm, 1e-12f) / 127.0f;
  }
}

// rest = resid - x_sub  (residual with outliers removed and masked columns zeroed)
__global__ void k_rest(const float* __restrict__ x, const float* __restrict__ colmask,
                       const float* __restrict__ scal, float* __restrict__ R) {
  float thr = scal[S_THR];
  int stride = gridDim.x * blockDim.x;
  for (int i = blockIdx.x * blockDim.x + threadIdx.x; i < NTOTX; i += stride) {
    __builtin_prefetch(x + i + 4 * stride, 0, 0);
    float v = x[i];
    float resid = (fabsf(v) > thr) ? 0.0f : v;
    R[i] = (colmask[i & (DIMX - 1)] != 0.0f) ? 0.0f : resid;
  }
}

static __device__ __forceinline__ unsigned int hash32(unsigned int v) {
  v ^= v >> 16; v *= 0x7feb352dU; v ^= v >> 15; v *= 0x846ca68bU; v ^= v >> 16;
  return v;
}

// Gaussian test matrix G (4096 x 16) via hash + Box-Muller
__global__ void k_gauss(float* __restrict__ G, int n) {
  int i = blockIdx.x * blockDim.x + threadIdx.x;
  if (i < n) {
    unsigned int h1 = hash32(0x9E3779B9u ^ (unsigned int)i);
    unsigned int h2 = hash32(0x85EBCA6Bu + (unsigned int)i);
    float u1 = ((h1 >> 8) + 1u) * (1.0f / 16777217.0f);
    float u2 = (h2 >> 8) * (1.0f / 16777216.0f);
    G[i] = sqrtf(-2.0f * logf(u1)) * cosf(6.28318530718f * u2);
  }
}

// Y(8192x16) = R(8192x4096) @ W(4096x16).
// Block = 4 waves = 64 rows. TDM stages a 64x32 fp32 tile of R into LDS,
// wave0 waits TENSORcnt, barrier, then each wave runs 8 WMMA K-steps on its
// own 16-row slab. A 16x4 f32 layout: lanes 0-15 M=lane K={0,1}; lanes 16-31
// M=lane-16 K={2,3}.
__global__ void k_gemm_rw(const float* __restrict__ R, const float* __restrict__ W,
                          float* __restrict__ Y) {
  __shared__ float tile[64 * BKX];
  int wave = threadIdx.x >> 5, lane = threadIdx.x & 31;
  int r0 = blockIdx.x * 64;
  int m = lane & 15, kh = lane >> 4, n = m;
  unsigned int lds0 = (unsigned int)(size_t)&tile[0];  // low 32 bits = LDS offset
  (void)lds0;
  v8f acc = {};
  for (int kk = 0; kk < DIMX; kk += BKX) {
#if HAVE_TDM
    if (wave == 0) {
      tdm_load_2d(lds0, R + (size_t)r0 * DIMX + kk, BKX, 64, DIMX, NROWSX, DIMX);
      __builtin_amdgcn_s_wait_tensorcnt(0);
    }
#else
    for (int t = threadIdx.x; t < 64 * BKX; t += blockDim.x) {
      int row = t / BKX, col = t - row * BKX;
      tile[t] = R[(size_t)(r0 + row) * DIMX + kk + col];
    }
#endif
    __syncthreads();
#pragma unroll
    for (int k4 = 0; k4 < BKX; k4 += 4) {
      v2f a = *(const v2f*)&tile[(wave * 16 + m) * BKX + k4 + 2 * kh];
      v2f b;
      b.x = W[(kk + k4 + 2 * kh) * RANKX + n];
      b.y = W[(kk + k4 + 2 * kh + 1) * RANKX + n];
      acc = wmma4(a, b, acc);
    }
    __syncthreads();
  }
#pragma unroll
  for (int v = 0; v < 8; ++v)
    Y[(r0 + wave * 16 + v + 8 * kh) * RANKX + n] = acc[v];
}

// Z(4096x16) = R^T(4096x8192) @ Q(8192x16).
// Block = 4 waves = 64 output columns of R. TDM stages a 32(row)x64(col) tile
// of R into LDS; A-fragment rows (fixed d, varying k) come from LDS columns.
__global__ void k_gemm_rtq(const float* __restrict__ R, const float* __restrict__ Qm,
                           float* __restrict__ Z) {
  __shared__ float tile[BKX * 64];
  int wave = threadIdx.x >> 5, lane = threadIdx.x & 31;
  int d0 = blockIdx.x * 64;
  int m = lane & 15, kh = lane >> 4, n = m;
  unsigned int lds0 = (unsigned int)(size_t)&tile[0];
  (void)lds0;
  v8f acc = {};
  for (int kk = 0; kk < NROWSX; kk += BKX) {
#if HAVE_TDM
    if (wave == 0) {
      tdm_load_2d(lds0, R + (size_t)kk * DIMX + d0, 64, BKX, DIMX, NROWSX, DIMX);
      __builtin_amdgcn_s_wait_tensorcnt(0);
    }
#else
    for (int t = threadIdx.x; t < BKX * 64; t += blockDim.x) {
      int row = t / 64, col = t - row * 64;
      tile[t] = R[(size_t)(kk + row) * DIMX + d0 + col];
    }
#endif
    __syncthreads();
#pragma unroll
    for (int k4 = 0; k4 < BKX; k4 += 4) {
      v2f a, b;
      a.x = tile[(k4 + 2 * kh) * 64 + wave * 16 + m];
      a.y = tile[(k4 + 2 * kh + 1) * 64 + wave * 16 + m];
      b.x = Qm[(kk + k4 + 2 * kh) * RANKX + n];
      b.y = Qm[(kk + k4 + 2 * kh + 1) * RANKX + n];
      acc = wmma4(a, b, acc);
    }
    __syncthreads();
  }
#pragma unroll
  for (int v = 0; v < 8; ++v)
    Z[(d0 + wave * 16 + v + 8 * kh) * RANKX + n] = acc[v];
}

// Gram(16x16) = W^T W over L rows (128 waves, WMMA accumulate + atomic merge)
__global__ void k_gram(const float* __restrict__ W, int L, float* __restrict__ Gram) {
  int wave = threadIdx.x >> 5, lane = threadIdx.x & 31;
  int w = blockIdx.x * 4 + wave;  // 128 waves total
  int m = lane & 15, kh = lane >> 4, n = m;
  v8f acc = {};
  for (int kk = w * 4; kk < L; kk += 128 * 4) {
    v2f a, b;
    a.x = W[(kk + 2 * kh) * RANKX + m];
    a.y = W[(kk + 2 * kh + 1) * RANKX + m];
    b.x = W[(kk + 2 * kh) * RANKX + n];
    b.y = W[(kk + 2 * kh + 1) * RANKX + n];
    acc = wmma4(a, b, acc);
  }
#pragma unroll
  for (int v = 0; v < 8; ++v)
    atomicAdd(&Gram[(v + 8 * kh) * 16 + n], acc[v]);
}

// Cholesky of Gram (G = U^T U) and upper-triangular inverse V = U^-1
__global__ void k_chol_inv(const float* __restrict__ Gram, float* __restrict__ Rinv) {
  if (threadIdx.x != 0 || blockIdx.x != 0) return;
  float A[16][16], U[16][16], V[16][16];
  for (int i = 0; i < 16; i++)
    for (int j = 0; j < 16; j++) {
      A[i][j] = Gram[i * 16 + j]; U[i][j] = 0.0f; V[i][j] = 0.0f;
    }
  for (int i = 0; i < 16; i++) {
    float d = A[i][i];
    for (int k = 0; k < i; k++) d -= U[k][i] * U[k][i];
    d = sqrtf(fmaxf(d, 1e-20f));
    U[i][i] = d;
    for (int j = i + 1; j < 16; j++) {
      float s = A[i][j];
      for (int k = 0; k < i; k++) s -= U[k][i] * U[k][j];
      U[i][j] = s / d;
    }
  }
  for (int i = 0; i < 16; i++) V[i][i] = 1.0f / U[i][i];
  for (int j = 1; j < 16; j++)
    for (int i = j - 1; i >= 0; i--) {
      float s = 0.0f;
      for (int k = i + 1; k <= j; k++) s += U[i][k] * V[k][j];
      V[i][j] = -s / U[i][i];
    }
  for (int i = 0; i < 16; i++)
    for (int j = 0; j < 16; j++) Rinv[i * 16 + j] = V[i][j];
}

// Q = Y * U^-1 (one row per thread, Rinv staged in LDS)
__global__ void k_apply_rinv(const float* __restrict__ Y, const float* __restrict__ Rinv,
                             float* __restrict__ Qout, int L) {
  __shared__ float rv[256];
  if (threadIdx.x < 256) rv[threadIdx.x] = Rinv[threadIdx.x];
  __syncthreads();
  int row = blockIdx.x * blockDim.x + threadIdx.x;
  if (row >= L) return;
  float y[16];
#pragma unroll
  for (int k = 0; k < 16; k++) y[k] = Y[row * 16 + k];
#pragma unroll
  for (int j = 0; j < 16; j++) {
    float s = 0.0f;
#pragma unroll
    for (int k = 0; k <= j; k++) s += y[k] * rv[k * 16 + j];
    Qout[row * 16 + j] = s;
  }
}

// Fused final pass: out_comp = x_out + x_sub_q + Q @ (Q^T R)
// low tile via K=16 WMMA; x re-read to reconstruct x_out / x_sub_q in-register.
__global__ void k_final(const float* __restrict__ x, const float* __restrict__ Q,
                        const float* __restrict__ Z, const float* __restrict__ colmask,
                        const float* __restrict__ scal, float* __restrict__ out_comp) {
  int wave = threadIdx.x >> 5, lane = threadIdx.x & 31;
  int wt = blockIdx.x * 4 + wave;  // 0..131071 wave tiles
  int r0 = (wt >> 8) << 4;         // 512 row tiles
  int d0 = (wt & 255) << 4;        // 256 col tiles
  int m = lane & 15, kh = lane >> 4, n = m;
  v8f acc = {};
#pragma unroll
  for (int kk = 0; kk < RANKX; kk += 4) {
    v2f a, b;
    a.x = Q[(r0 + m) * RANKX + kk + 2 * kh];
    a.y = Q[(r0 + m) * RANKX + kk + 2 * kh + 1];
    b.x = Z[(d0 + n) * RANKX + kk + 2 * kh];   // B[k][n] = Z[d0+n][k]
    b.y = Z[(d0 + n) * RANKX + kk + 2 * kh + 1];
    acc = wmma4(a, b, acc);
  }
  float thr = scal[S_THR], scale = scal[S_SCALE];
  float cm = colmask[d0 + n];
#pragma unroll
  for (int v = 0; v < 8; ++v) {
    int idx = (r0 + v + 8 * kh) * DIMX + d0 + n;
    float xv = x[idx];
    bool isout = fabsf(xv) > thr;
    float resid = isout ? 0.0f : xv;
    float xout = isout ? xv : 0.0f;
    float q = 0.0f;
    if (cm != 0.0f)
      q = fminf(fmaxf(roundf(resid / scale), -128.0f), 127.0f) * scale;
    out_comp[idx] = xout + q + acc[v];
  }
}

extern "C" void kernel_launch(void* const* d_in, const int* in_sizes, int n_in,
                              void* d_out, int out_size, void* d_ws, size_t ws_size,
                              hipStream_t stream) {
  (void)in_sizes; (void)n_in; (void)out_size; (void)ws_size;
  const float* x = (const float*)d_in[0];
  float* dout = (float*)d_out;
  float* out_res  = dout;              // silu(x), NTOTX floats
  float* out_thr  = dout + NTOTX;      // scalar quantile threshold
  float* out_comp = dout + NTOTX + 1;  // x_compressed, NTOTX floats

  float* ws    = (float*)d_ws;
  float* R     = ws;                         // NTOTX (residual matrix "rest")
  float* G     = R + NTOTX;                  // 4096x16
  float* Yb    = G + DIMX * RANKX;           // 8192x16
  float* Qb    = Yb + NROWSX * RANKX;        // 8192x16
  float* Zb    = Qb + NROWSX * RANKX;        // 4096x16
  float* Q2b   = Zb + DIMX * RANKX;          // 4096x16
  float* Gram  = Q2b + DIMX * RANKX;         // 16x16
  float* Rinv  = Gram + 256;                 // 16x16
  float* cn2   = Rinv + 256;                 // 4096 column sumsq
  float* cmask = cn2 + DIMX;                 // 4096 column mask
  unsigned int* h1 = (unsigned int*)(cmask + DIMX);  // 4096 bins
  unsigned int* h2 = h1 + NBINSX;                    // 4096 bins
  float* scal = (float*)(h2 + NBINSX);               // 64 scalar slots

  int zn = 256 + 256 + DIMX + DIMX + NBINSX + NBINSX + 64;
  k_zero<<<(zn + 255) / 256, 256, 0, stream>>>(Gram, zn);

  // silu + column norms + absmax
  k_silu_stats<<<NTOTX / 65536, 256, 0, stream>>>(x, out_res, cn2, scal);
  // top-k columns
  k_topk<<<1, 1024, 0, stream>>>(cn2, cmask);
  // two-level histogram quantile
  k_prep1<<<1, 32, 0, stream>>>(scal);
  k_hist<<<1024, 256, 0, stream>>>(x, scal, h1);
  k_scan<<<1, 32, 0, stream>>>(h1, scal, 0, nullptr);
  k_hist<<<1024, 256, 0, stream>>>(x, scal, h2);
  k_scan<<<1, 32, 0, stream>>>(h2, scal, 1, out_thr);
  // int8 fake-quant scale for sub-outlier columns
  k_subamax<<<1024, 256, 0, stream>>>(x, cmask, scal);
  k_scale<<<1, 32, 0, stream>>>(scal);
  // residual for the low-rank stage
  k_rest<<<1024, 256, 0, stream>>>(x, cmask, scal, R);
  // random test matrix
  k_gauss<<<(DIMX * RANKX + 255) / 256, 256, 0, stream>>>(G, DIMX * RANKX);

  auto qr = [&](float* M, int L, float* Qout) {
    k_zero<<<1, 256, 0, stream>>>(Gram, 256);
    k_gram<<<32, 128, 0, stream>>>(M, L, Gram);
    k_chol_inv<<<1, 32, 0, stream>>>(Gram, Rinv);
    k_apply_rinv<<<(L + 255) / 256, 256, 0, stream>>>(M, Rinv, Qout, L);
  };

  // RSVD power iteration (SVD of Bm collapses to low = Q Q^T R)
  k_gemm_rw<<<NROWSX / 64, 128, 0, stream>>>(R, G, Yb);
  qr(Yb, NROWSX, Qb);
  for (int it = 0; it < 2; ++it) {
    k_gemm_rtq<<<DIMX / 64, 128, 0, stream>>>(R, Qb, Zb);
    qr(Zb, DIMX, Q2b);
    k_gemm_rw<<<NROWSX / 64, 128, 0, stream>>>(R, Q2b, Yb);
    qr(Yb, NROWSX, Qb);
  }
  k_gemm_rtq<<<DIMX / 64, 128, 0, stream>>>(R, Qb, Zb);  // Zb = (Q^T R)^T

  // fused: x_out + x_sub_q + Q @ Z^T  -> out_comp
  k_final<<<(NROWSX / 16) * (DIMX / 16) / 4, 128, 0, stream>>>(x, Qb, Zb, cmask,
                                                               scal, out_comp);
}